// ConcatScore_73315091742826
// MI455X (gfx1250) — compile-verified
//
#include <hip/hip_runtime.h>
#include <hip/hip_bf16.h>

// Problem constants (from reference): B=8, S=64, T=30, D_W=400, D_T=20, R=256
#define BS   512          // B*S
#define DW   400
#define DT   20
#define TT   30
#define RR   256
#define FAN  440          // D_W + 2*D_T  (W row stride)
#define NOUT (TT * TT)    // 900 outputs per bs row

typedef float v2f __attribute__((ext_vector_type(2)));
typedef float v8f __attribute__((ext_vector_type(8)));

__device__ __forceinline__ float fast_tanh(float x) {
#if __has_builtin(__builtin_amdgcn_tanhf)
    return __builtin_amdgcn_tanhf(x);
#else
    float y;
    asm volatile("v_tanh_f32 %0, %1" : "=v"(y) : "v"(x));
    return y;
#endif
}

// ---------------------------------------------------------------------------
// Kernel A: word_part[bs][r] = sum_d word[bs][d] * W[r][d]   (d < 400)
// One wave (32 lanes) per 16x16 tile of the 512x256 result.
// V_WMMA_F32_16X16X4_F32: A 16x4 (M=lane&15, K pair selected by lane half),
// B 4x16 (mirror layout), C/D 16x16 with M = vgpr + 8*(lane>>4), N = lane&15.
// ---------------------------------------------------------------------------
__global__ void __launch_bounds__(32)
wordpart_wmma_kernel(const float* __restrict__ word,
                     const float* __restrict__ W,
                     float* __restrict__ wp) {
    const int m0   = blockIdx.x * 16;   // bs tile  (32 tiles)
    const int n0   = blockIdx.y * 16;   // r tile   (16 tiles)
    const int lane = threadIdx.x;
    const int half = lane >> 4;         // 0 or 1
    const int mn   = lane & 15;

    const float* arow = word + (size_t)(m0 + mn) * DW + 2 * half;
    const float* brow = W + (size_t)(n0 + mn) * FAN + 2 * half;

    v8f acc = {0.f, 0.f, 0.f, 0.f, 0.f, 0.f, 0.f, 0.f};

    for (int k = 0; k < DW; k += 4) {
        v2f a, b;
        a.x = arow[k];
        a.y = arow[k + 1];
        b.x = brow[k];
        b.y = brow[k + 1];
        acc = __builtin_amdgcn_wmma_f32_16x16x4_f32(
            false, a, false, b, (short)0, acc, false, false);
    }

    float* dcol = wp + n0 + mn;  // column n = n0 + mn
#pragma unroll
    for (int v = 0; v < 8; ++v) {
        const int m = v + 8 * half;
        dcol[(size_t)(m0 + m) * RR] = acc[v];
    }
}

// ---------------------------------------------------------------------------
// Kernel B: tag_j[t][r] = tag[t]·W[r][400:420],
//           tag_i[t][r] = tag[t]·W[r][420:440] + b[r]   (bias folded in)
// ---------------------------------------------------------------------------
__global__ void __launch_bounds__(RR)
tag_kernel(const float* __restrict__ tag,
           const float* __restrict__ W,
           const float* __restrict__ bias,
           float* __restrict__ ti,
           float* __restrict__ tj) {
    const int t = blockIdx.x;
    const int r = threadIdx.x;
    const float* te = tag + t * DT;
    const float* w1 = W + (size_t)r * FAN + DW;        // -> tag_j
    const float* w2 = W + (size_t)r * FAN + DW + DT;   // -> tag_i
    float aj = 0.f, ai = 0.f;
#pragma unroll
    for (int d = 0; d < DT; ++d) {
        const float x = te[d];
        aj += x * w1[d];
        ai += x * w2[d];
    }
    tj[t * RR + r] = aj;
    ti[t * RR + r] = ai + bias[r];
}

// ---------------------------------------------------------------------------
// Kernel C (hot): score[bs][i][j] = sum_r v[r]*tanh(wp[bs][r]+ti[i][r]+tj[j][r])
// One block per bs row; operands staged in LDS (63.5 KB / block).
// Loop nest inverted vs naive: each thread carries 4 output accumulators so
// wp4/v4 LDS reads are shared across its 4 outputs -> ~2.5 LDS dwords per
// tanh instead of 4, keeping LDS under the v_tanh_f32 TRANS-pipe rate.
// ---------------------------------------------------------------------------
__global__ void __launch_bounds__(RR)
score_kernel(const float* __restrict__ wp,
             const float* __restrict__ ti,
             const float* __restrict__ tj,
             const float* __restrict__ vec,
             float* __restrict__ out) {
    __shared__ float s_wp[RR];
    __shared__ float s_v[RR];
    __shared__ float s_ti[TT * RR];
    __shared__ float s_tj[TT * RR];

    const int bs  = blockIdx.x;
    const int tid = threadIdx.x;   // 256 threads

    s_wp[tid] = wp[(size_t)bs * RR + tid];
    s_v[tid]  = vec[tid];
    for (int idx = tid; idx < TT * RR; idx += RR) {
        s_ti[idx] = ti[idx];
        s_tj[idx] = tj[idx];
    }
    __syncthreads();

    // This thread's 4 outputs (4th clamped; store guarded below, no race).
    const int  o0   = tid;
    const int  o1   = tid + RR;
    const int  o2   = tid + 2 * RR;
    const bool has3 = (tid + 3 * RR) < NOUT;
    const int  o3   = has3 ? (tid + 3 * RR) : (NOUT - 1);

    const float4* p_wp = (const float4*)s_wp;
    const float4* p_v  = (const float4*)s_v;
    const float4* ti0  = (const float4*)(s_ti + (o0 / TT) * RR);
    const float4* tj0  = (const float4*)(s_tj + (o0 % TT) * RR);
    const float4* ti1  = (const float4*)(s_ti + (o1 / TT) * RR);
    const float4* tj1  = (const float4*)(s_tj + (o1 % TT) * RR);
    const float4* ti2  = (const float4*)(s_ti + (o2 / TT) * RR);
    const float4* tj2  = (const float4*)(s_tj + (o2 % TT) * RR);
    const float4* ti3  = (const float4*)(s_ti + (o3 / TT) * RR);
    const float4* tj3  = (const float4*)(s_tj + (o3 % TT) * RR);

    float acc0 = 0.f, acc1 = 0.f, acc2 = 0.f, acc3 = 0.f;

#pragma unroll 2
    for (int rq = 0; rq < RR / 4; ++rq) {
        const float4 w4 = p_wp[rq];   // shared across the 4 outputs
        const float4 v4 = p_v[rq];

        float4 a, b;
        a = ti0[rq]; b = tj0[rq];
        acc0 += v4.x * fast_tanh(w4.x + a.x + b.x);
        acc0 += v4.y * fast_tanh(w4.y + a.y + b.y);
        acc0 += v4.z * fast_tanh(w4.z + a.z + b.z);
        acc0 += v4.w * fast_tanh(w4.w + a.w + b.w);

        a = ti1[rq]; b = tj1[rq];
        acc1 += v4.x * fast_tanh(w4.x + a.x + b.x);
        acc1 += v4.y * fast_tanh(w4.y + a.y + b.y);
        acc1 += v4.z * fast_tanh(w4.z + a.z + b.z);
        acc1 += v4.w * fast_tanh(w4.w + a.w + b.w);

        a = ti2[rq]; b = tj2[rq];
        acc2 += v4.x * fast_tanh(w4.x + a.x + b.x);
        acc2 += v4.y * fast_tanh(w4.y + a.y + b.y);
        acc2 += v4.z * fast_tanh(w4.z + a.z + b.z);
        acc2 += v4.w * fast_tanh(w4.w + a.w + b.w);

        a = ti3[rq]; b = tj3[rq];
        acc3 += v4.x * fast_tanh(w4.x + a.x + b.x);
        acc3 += v4.y * fast_tanh(w4.y + a.y + b.y);
        acc3 += v4.z * fast_tanh(w4.z + a.z + b.z);
        acc3 += v4.w * fast_tanh(w4.w + a.w + b.w);
    }

    float* orow = out + (size_t)bs * NOUT;
    orow[o0] = acc0;
    orow[o1] = acc1;
    orow[o2] = acc2;
    if (has3) orow[o3] = acc3;
}

// ---------------------------------------------------------------------------
extern "C" void kernel_launch(void* const* d_in, const int* in_sizes, int n_in,
                              void* d_out, int out_size, void* d_ws, size_t ws_size,
                              hipStream_t stream) {
    (void)in_sizes; (void)n_in; (void)out_size; (void)ws_size;

    const float* word = (const float*)d_in[0];   // (512, 400)
    const float* tag  = (const float*)d_in[1];   // (30, 20)
    const float* W    = (const float*)d_in[2];   // (256, 440)
    const float* bias = (const float*)d_in[3];   // (256,)
    const float* vec  = (const float*)d_in[4];   // (1, 256)
    float* out = (float*)d_out;                  // (8,64,30,30,1)

    // Workspace layout
    float* wp = (float*)d_ws;                    // 512*256 f32 = 512 KB
    float* ti = wp + (size_t)BS * RR;            // 30*256 f32
    float* tj = ti + (size_t)TT * RR;            // 30*256 f32

    wordpart_wmma_kernel<<<dim3(BS / 16, RR / 16), 32, 0, stream>>>(word, W, wp);
    tag_kernel<<<TT, RR, 0, stream>>>(tag, W, bias, ti, tj);
    score_kernel<<<BS, RR, 0, stream>>>(wp, ti, tj, vec, out);
}